// HeteroRGCN_61735859913388
// MI455X (gfx1250) — compile-verified
//
#include <hip/hip_runtime.h>

// HeteroRGCN forward for MI455X (gfx1250, wave32).
// Dead-code analysis: final output depends only on
//   features -> (W0_t2c / W0_t2m GEMMs) -> seg-mean over t2c/t2m -> leaky
//            -> (W1_c2t / W1_m2t GEMMs) -> seg-mean over c2t/m2t -> Wf.
// emb_client / emb_merchant / W0_c2t / W0_m2t / W1_t2c / W1_t2m never reach h1["t"].

#define NT_N 500000
#define NC_N 100000
#define NM_N 20000
#define NEDGE 500000
#define HID 64

typedef float v2f __attribute__((ext_vector_type(2)));
typedef float v8f __attribute__((ext_vector_type(8)));

__global__ void zero_kernel(float* __restrict__ p, size_t n) {
    size_t i = (size_t)blockIdx.x * blockDim.x + threadIdx.x;
    size_t stride = (size_t)gridDim.x * blockDim.x;
    for (; i < n; i += stride) p[i] = 0.0f;
}

__global__ void count_kernel(const int* __restrict__ dst, float* __restrict__ cnt, int E) {
    int e = blockIdx.x * 256 + threadIdx.x;
    if (e < E) atomicAdd(&cnt[dst[e]], 1.0f);
}

__global__ void recip_kernel(float* __restrict__ c, int n) {
    int i = blockIdx.x * 256 + threadIdx.x;
    if (i < n) c[i] = 1.0f / fmaxf(c[i], 1.0f);
}

// One wave covers half an edge-row (32 of 64 feats): gather wh[src], scale by
// 1/deg(dst), float-atomic scatter into acc[dst]. Sum of scaled messages over
// all relations sharing `acc` == sum of per-relation segment means.
__global__ void scatter_kernel(const float* __restrict__ wh, const int* __restrict__ src,
                               const int* __restrict__ dst, const float* __restrict__ inv,
                               float* __restrict__ acc, int E) {
    int t = blockIdx.x * 256 + threadIdx.x;
    int e = t >> 6;
    int f = t & 63;
    if (e < E) {
        int s = src[e];
        int d = dst[e];
        float v = wh[(size_t)s * HID + f] * inv[d];
        atomicAdd(&acc[(size_t)d * HID + f], v);
    }
}

// out[nrows x 64] = (LEAKY? leaky_relu(A) : A)[nrows x K] @ W[K x 64] + b, for
// NMAT weight matrices sharing the same A (fused A traffic for layer 0).
// Wave = 16 rows x 64 cols = 4 accumulator tiles per matrix, K-step of 4 via
// v_wmma_f32_16x16x4_f32. W staged in LDS pre-swizzled into B-fragment order
// so each fragment is one conflict-free ds_load_b64 per lane.
template <int K, int NMAT, bool LEAKY>
__launch_bounds__(256)
__global__ void rgcn_gemm(const float* __restrict__ A, int nrows,
                          const float* __restrict__ W0, const float* __restrict__ b0,
                          float* __restrict__ out0,
                          const float* __restrict__ W1, const float* __restrict__ b1,
                          float* __restrict__ out1) {
    __shared__ float lds[NMAT * K * 64];

    // Cooperative fill in B-fragment order:
    // frag(s, n) = 64 words; word = lane*2 + d; value = W[K= 4s + (lane<16? d : 2+d)][n*16 + lane%16]
    for (int w = threadIdx.x; w < NMAT * K * 64; w += 256) {
        int m      = w / (K * 64);
        int w2     = w - m * (K * 64);
        int fragId = w2 >> 6;
        int within = w2 & 63;
        int ln     = within >> 1;
        int d      = within & 1;
        int s      = fragId >> 2;
        int n      = fragId & 3;
        int kk     = 4 * s + ((ln < 16) ? d : 2 + d);
        int col    = (n << 4) + (ln & 15);
        const float* Wp = (m == 0) ? W0 : W1;
        lds[w] = Wp[kk * 64 + col];
    }
    __syncthreads();

    const int lane = threadIdx.x & 31;
    const int wave = threadIdx.x >> 5;
    const int m0   = blockIdx.x * 128 + wave * 16;

    // A fragment: lane L holds row m0 + L%16; v0/v1 = K = 4s+{0,1} (lanes 0-15)
    // or 4s+{2,3} (lanes 16-31). Out-of-range rows clamp (results discarded).
    int rowA = m0 + (lane & 15);
    int rowc = rowA < nrows ? rowA : (nrows - 1);
    const float* aptr = A + (size_t)rowc * K + ((lane & 16) ? 2 : 0);

    v8f acc[NMAT][4];
#pragma unroll
    for (int m = 0; m < NMAT; ++m)
#pragma unroll
        for (int n = 0; n < 4; ++n) {
            v8f z = {0.f, 0.f, 0.f, 0.f, 0.f, 0.f, 0.f, 0.f};
            acc[m][n] = z;
        }

#pragma unroll 4
    for (int s = 0; s < K / 4; ++s) {
        v2f a = *(const v2f*)(aptr + 4 * s);
        if (LEAKY) {
            a.x = (a.x > 0.f) ? a.x : 0.01f * a.x;
            a.y = (a.y > 0.f) ? a.y : 0.01f * a.y;
        }
#pragma unroll
        for (int m = 0; m < NMAT; ++m)
#pragma unroll
            for (int n = 0; n < 4; ++n) {
                v2f b = *(const v2f*)(&lds[m * (K * 64) + ((s * 4 + n) << 6) + lane * 2]);
                acc[m][n] = __builtin_amdgcn_wmma_f32_16x16x4_f32(
                    false, a, false, b, (short)0, acc[m][n], false, false);
            }
    }

    // C/D layout: VGPR r -> M = r (lanes 0-15) / r+8 (lanes 16-31), N = lane%16.
    const int colBase = lane & 15;
    const int rowOff  = (lane >> 4) << 3;
#pragma unroll
    for (int m = 0; m < NMAT; ++m) {
        float* op        = (m == 0) ? out0 : out1;
        const float* bp  = (m == 0) ? b0 : b1;
#pragma unroll
        for (int r = 0; r < 8; ++r) {
            int row = m0 + rowOff + r;
            if (row < nrows) {
#pragma unroll
                for (int n = 0; n < 4; ++n)
                    op[(size_t)row * 64 + (n << 4) + colBase] =
                        acc[m][n][r] + bp[(n << 4) + colBase];
            }
        }
    }
}

// out[i, 0:2] = acc_t[i, 0:64] @ Wf(64x2) + bf
__global__ void final_kernel(const float* __restrict__ acc, const float* __restrict__ Wf,
                             const float* __restrict__ bf, float* __restrict__ out, int n) {
    __shared__ float w[128];
    __shared__ float b2[2];
    if (threadIdx.x < 128) w[threadIdx.x] = Wf[threadIdx.x];
    if (threadIdx.x < 2) b2[threadIdx.x] = bf[threadIdx.x];
    __syncthreads();
    int i = blockIdx.x * 256 + threadIdx.x;
    if (i < n) {
        float s0 = b2[0], s1 = b2[1];
        const float* a = acc + (size_t)i * 64;
#pragma unroll
        for (int k = 0; k < 64; ++k) {
            float v = a[k];
            s0 += v * w[2 * k];
            s1 += v * w[2 * k + 1];
        }
        out[2 * i]     = s0;
        out[2 * i + 1] = s1;
    }
}

extern "C" void kernel_launch(void* const* d_in, const int* in_sizes, int n_in,
                              void* d_out, int out_size, void* d_ws, size_t ws_size,
                              hipStream_t stream) {
    (void)in_sizes; (void)n_in; (void)out_size; (void)ws_size;

    const float* features = (const float*)d_in[0];
    const int* src_c2t = (const int*)d_in[3];
    const int* dst_c2t = (const int*)d_in[4];
    const int* src_m2t = (const int*)d_in[5];
    const int* dst_m2t = (const int*)d_in[6];
    const int* src_t2c = (const int*)d_in[7];
    const int* dst_t2c = (const int*)d_in[8];
    const int* src_t2m = (const int*)d_in[9];
    const int* dst_t2m = (const int*)d_in[10];
    const float* W1_c2t = (const float*)d_in[13];
    const float* b1_c2t = (const float*)d_in[14];
    const float* W1_m2t = (const float*)d_in[17];
    const float* b1_m2t = (const float*)d_in[18];
    const float* W0_t2c = (const float*)d_in[19];
    const float* b0_t2c = (const float*)d_in[20];
    const float* W0_t2m = (const float*)d_in[23];
    const float* b0_t2m = (const float*)d_in[24];
    const float* Wf = (const float*)d_in[27];
    const float* bf = (const float*)d_in[28];
    float* out = (float*)d_out;

    // Workspace layout (floats). whA doubles as layer-1 acc_t after its
    // consumers (t2c scatter) are done. Total ~322 MB.
    float* ws = (float*)d_ws;
    size_t o = 0;
    float* whA    = ws + o; o += (size_t)NT_N * HID;   // wh0_t2c, later acc_t
    float* whB    = ws + o; o += (size_t)NT_N * HID;   // wh0_t2m
    float* zbase  = ws + o;                            // start of zero region
    float* acc_c  = ws + o; o += (size_t)NC_N * HID;
    float* acc_m  = ws + o; o += (size_t)NM_N * HID;
    float* wh1c   = ws + o; o += (size_t)NC_N * HID;
    float* wh1m   = ws + o; o += (size_t)NM_N * HID;
    float* cnt_c  = ws + o; o += NC_N;
    float* cnt_m  = ws + o; o += NM_N;
    float* cnt_tc = ws + o; o += NT_N;
    float* cnt_tm = ws + o; o += NT_N;
    size_t zcount = (size_t)((ws + o) - zbase);

    zero_kernel<<<2048, 256, 0, stream>>>(zbase, zcount);

    int egrid = (NEDGE + 255) / 256;
    count_kernel<<<egrid, 256, 0, stream>>>(dst_t2c, cnt_c, NEDGE);
    count_kernel<<<egrid, 256, 0, stream>>>(dst_t2m, cnt_m, NEDGE);
    count_kernel<<<egrid, 256, 0, stream>>>(dst_c2t, cnt_tc, NEDGE);
    count_kernel<<<egrid, 256, 0, stream>>>(dst_m2t, cnt_tm, NEDGE);
    recip_kernel<<<(NC_N + 255) / 256, 256, 0, stream>>>(cnt_c, NC_N);
    recip_kernel<<<(NM_N + 255) / 256, 256, 0, stream>>>(cnt_m, NM_N);
    recip_kernel<<<(NT_N + 255) / 256, 256, 0, stream>>>(cnt_tc, NT_N);
    recip_kernel<<<(NT_N + 255) / 256, 256, 0, stream>>>(cnt_tm, NT_N);

    // Layer 0: both 500Kx128x64 GEMMs fused (shared A-fragment loads).
    rgcn_gemm<128, 2, false><<<(NT_N + 127) / 128, 256, 0, stream>>>(
        features, NT_N, W0_t2c, b0_t2c, whA, W0_t2m, b0_t2m, whB);

    int sgrid = (int)(((size_t)NEDGE * 64 + 255) / 256);
    scatter_kernel<<<sgrid, 256, 0, stream>>>(whA, src_t2c, dst_t2c, cnt_c, acc_c, NEDGE);
    scatter_kernel<<<sgrid, 256, 0, stream>>>(whB, src_t2m, dst_t2m, cnt_m, acc_m, NEDGE);

    // whA's data fully consumed; reuse as acc_t.
    zero_kernel<<<2048, 256, 0, stream>>>(whA, (size_t)NT_N * HID);

    // Layer 1 (leaky_relu fused into A-load).
    rgcn_gemm<64, 1, true><<<(NC_N + 127) / 128, 256, 0, stream>>>(
        acc_c, NC_N, W1_c2t, b1_c2t, wh1c, nullptr, nullptr, nullptr);
    rgcn_gemm<64, 1, true><<<(NM_N + 127) / 128, 256, 0, stream>>>(
        acc_m, NM_N, W1_m2t, b1_m2t, wh1m, nullptr, nullptr, nullptr);

    scatter_kernel<<<sgrid, 256, 0, stream>>>(wh1c, src_c2t, dst_c2t, cnt_tc, whA, NEDGE);
    scatter_kernel<<<sgrid, 256, 0, stream>>>(wh1m, src_m2t, dst_m2t, cnt_tm, whA, NEDGE);

    final_kernel<<<(NT_N + 255) / 256, 256, 0, stream>>>(whA, Wf, bf, out, NT_N);
}